// CellSampler_78426102825150
// MI455X (gfx1250) — compile-verified
//
#include <hip/hip_runtime.h>
#include <stdint.h>

#define BATCH   256
#define NROWS   4096
#define DIM     64
#define MAXF    128
#define THREADS 256
#define NWAVES  (THREADS / 32)
#define CONF    0.7f

__global__ __launch_bounds__(THREADS) void cell_sampler_kernel(
    const float* __restrict__ feats,
    float* __restrict__ out_feat,
    float* __restrict__ out_mask) {
  __shared__ int    s_src[MAXF];                 // selected source row indices
  __shared__ int    s_wcnt[NWAVES];              // per-wave hit counts
  __shared__ float4 s_stage[MAXF * (DIM / 4)];   // 128 rows * 256B = 32 KB

  const int tid  = threadIdx.x;
  const int lane = tid & 31;
  const int wave = tid >> 5;
  const int b    = blockIdx.x;
  const float* batch = feats + (size_t)b * NROWS * DIM;

  // ---------------- Phase 1: ballot-based block-wide stream compaction ------
  int base = 0;  // uniform running count of selected rows
  for (int n0 = 0; n0 < NROWS; n0 += THREADS) {
    if (base >= MAXF) break;  // uniform early exit
    const int n = n0 + tid;
    const float conf = batch[(size_t)n * DIM + 9];
    const bool m = conf > CONF;

    unsigned bal = __builtin_amdgcn_ballot_w32(m);     // wave32 ballot
    int lanePre  = __popc(bal & ((1u << lane) - 1u));  // intra-wave excl. prefix
    if (lane == 0) s_wcnt[wave] = __popc(bal);
    __syncthreads();

    int wbase = base;
    for (int w = 0; w < wave; ++w) wbase += s_wcnt[w];
    const int pos = wbase + lanePre;
    if (m && pos < MAXF) s_src[pos] = n;

    int total = 0;
    for (int w = 0; w < NWAVES; ++w) total += s_wcnt[w];
    base += total;
    __syncthreads();  // protect s_wcnt reuse + s_src visibility
  }
  const int count = base < MAXF ? base : MAXF;
  __syncthreads();

  // ---------------- Phase 2: async DMA gather of selected rows into LDS -----
  // 128 rows * 16 chunks of 16B = 2048 chunk tasks; 8 per thread.
  const int chunks = MAXF * (DIM / 4);  // 2048 float4 chunks
  for (int g = tid; g < chunks; g += THREADS) {
    const int r = g >> 4;   // row slot
    const int c = g & 15;   // 16B chunk within row
    if (r < count) {
      const float* src = batch + (size_t)s_src[r] * DIM + c * 4;
      unsigned lds_off = (unsigned)(uintptr_t)(&s_stage[g]);  // low 32b = LDS offset
      unsigned long long ga = (unsigned long long)(uintptr_t)src;
      // CDNA5 async global->LDS DMA (ASYNCcnt-tracked), 16B per active lane
      asm volatile("global_load_async_to_lds_b128 %0, %1, off"
                   :: "v"(lds_off), "v"(ga)
                   : "memory");
    }
  }
  // wait for this wave's async DMA, then barrier so all staged data is visible
  asm volatile("s_wait_asynccnt 0" ::: "memory");
  __syncthreads();

  // ---------------- Phase 3: write compacted rows (zero-fill the tail) ------
  float4* outv = (float4*)(out_feat + (size_t)b * MAXF * DIM);
  const float4 z = {0.f, 0.f, 0.f, 0.f};
  for (int g = tid; g < chunks; g += THREADS) {
    const int r = g >> 4;
    outv[g] = (r < count) ? s_stage[g] : z;  // ds_load_b128 -> global_store_b128
  }

  if (tid < MAXF)
    out_mask[(size_t)b * MAXF + tid] = (tid < count) ? 1.0f : 0.0f;
}

extern "C" void kernel_launch(void* const* d_in, const int* in_sizes, int n_in,
                              void* d_out, int out_size, void* d_ws, size_t ws_size,
                              hipStream_t stream) {
  (void)in_sizes; (void)n_in; (void)out_size; (void)d_ws; (void)ws_size;
  const float* feats = (const float*)d_in[0];
  float* out_feat = (float*)d_out;                              // [256,128,64]
  float* out_mask = out_feat + (size_t)BATCH * MAXF * DIM;      // [256,128]
  cell_sampler_kernel<<<BATCH, THREADS, 0, stream>>>(feats, out_feat, out_mask);
}